// GAT_10548439679259
// MI455X (gfx1250) — compile-verified
//
#include <hip/hip_runtime.h>
#include <math.h>

#define NN   1600
#define ROI  200
#define HID  64
#define HIN  8
#define D1   512
#define EE   320000
#define BB   8
#define NEG  0.2f

typedef __attribute__((ext_vector_type(2))) float v2f;
typedef __attribute__((ext_vector_type(8))) float v8f;

// ---- order-preserving float <-> uint encoding for atomicMax on floats ----
__device__ __forceinline__ unsigned enc_f(float x) {
    unsigned u = __float_as_uint(x);
    return (u & 0x80000000u) ? ~u : (u | 0x80000000u);
}
__device__ __forceinline__ float dec_f(unsigned u) {
    return (u & 0x80000000u) ? __uint_as_float(u & 0x7fffffffu)
                             : __uint_as_float(~u);
}

// ---------------------------------------------------------------------------
// LDS-staged f32 WMMA GEMM: C[M,N] = A[M,K] @ B[K,N], row-major.
// Block = 4 waves, each wave owns a 16 x (NT*16) slab of C (NT accumulators).
// The block's 16 x KT A-strip is staged in LDS (coalesced global loads,
// padded stride to avoid bank conflicts), reused by all 4 waves and all NT
// column tiles -> 4*NT WMMA per LDS fragment. B rows are read directly from
// global, coalesced in 64B segments (L2-resident operands).
// Requires M%16==0, K%4==0; launch must cover N with grid.x*4*NT*16 >= N and
// N % (NT*16) == 0.
// V_WMMA_F32_16X16X4_F32 lane layouts (CDNA5 ISA 7.12.2):
//   A 16x4 : lane m=lane&15, holds K = kb, kb+1 with kb = 2*(lane>=16)
//   B 4x16 : lane n=lane&15, holds rows kb, kb+1
//   C 16x16: VGPR v -> row v + 8*(lane>=16), col lane&15
// ---------------------------------------------------------------------------
#define KT 128
#define STRIDE 132

template <int NT>
__global__ void k_wmma_gemm_f32(const float* __restrict__ A,
                                const float* __restrict__ B,
                                float* __restrict__ C,
                                int M, int N, int K, int lda, int ldb, int ldc) {
    __shared__ float smA[16 * STRIDE];
    const int tid  = threadIdx.x;
    const int lane = tid & 31;
    const int wv   = tid >> 5;
    const int m0   = blockIdx.y * 16;
    const int nw   = (blockIdx.x * 4 + wv) * (NT * 16);
    const bool active = (nw < N);            // wave-uniform
    const int l15 = lane & 15;
    const int kb  = (lane >> 4) << 1;        // 0 or 2

    v8f zero = {};
    v8f acc[NT];
#pragma unroll
    for (int t = 0; t < NT; ++t) acc[t] = zero;

    const float* Bp[NT];
#pragma unroll
    for (int t = 0; t < NT; ++t)
        Bp[t] = B + (size_t)kb * ldb + (nw + t * 16 + l15);

    for (int k0 = 0; k0 < K; k0 += KT) {
        const int chunk = (K - k0) < KT ? (K - k0) : KT;
        __syncthreads();
        // cooperative, coalesced staging of the 16 x chunk A-strip
        for (int i = tid; i < 16 * chunk; i += blockDim.x) {
            int r = i / chunk;
            int c = i - r * chunk;
            smA[r * STRIDE + c] = A[(size_t)(m0 + r) * lda + (k0 + c)];
        }
        __syncthreads();
        if (active) {
            const float* arow = smA + l15 * STRIDE + kb;
            for (int kk = 0; kk < chunk; kk += 4) {
                v2f a;
                a[0] = arow[kk];
                a[1] = arow[kk + 1];
#pragma unroll
                for (int t = 0; t < NT; ++t) {
                    v2f b;
                    b[0] = Bp[t][(size_t)(k0 + kk) * ldb];
                    b[1] = Bp[t][(size_t)(k0 + kk + 1) * ldb];
                    acc[t] = __builtin_amdgcn_wmma_f32_16x16x4_f32(
                        /*neg_a=*/false, a, /*neg_b=*/false, b,
                        /*c_mod=*/(short)0, acc[t],
                        /*reuse_a=*/false, /*reuse_b=*/false);
                }
            }
        }
    }
    if (active) {
        const int row = m0 + ((lane >> 4) << 3);
#pragma unroll
        for (int t = 0; t < NT; ++t) {
            float* Cp = C + (size_t)row * ldc + (nw + t * 16 + l15);
#pragma unroll
            for (int v = 0; v < 8; ++v) Cp[(size_t)v * ldc] = acc[t][v];
        }
    }
}

// ---------------- attention coefficients: asrc/adst[n,h] = <h[n,h,:], a> ----
__global__ void k_attn_coef(const float* __restrict__ h,
                            const float* __restrict__ a_src,
                            const float* __restrict__ a_dst,
                            float* __restrict__ asrc, float* __restrict__ adst,
                            int H, int F) {
    int i = blockIdx.x * blockDim.x + threadIdx.x;   // i = n*H + hh
    if (i >= NN * H) return;
    int hh = i % H;
    int n  = i / H;
    const float* hp = h + (size_t)n * H * F + (size_t)hh * F;
    const float* as = a_src + (size_t)hh * F;
    const float* ad = a_dst + (size_t)hh * F;
    float s = 0.f, d = 0.f;
    for (int f = 0; f < F; ++f) { float v = hp[f]; s += v * as[f]; d += v * ad[f]; }
    asrc[i] = s;
    adst[i] = d;
}

// ---------------- zero fills ----------------
__global__ void k_zero(float* p, int n) {
    int i = blockIdx.x * blockDim.x + threadIdx.x;
    if (i < n) p[i] = 0.f;
}
__global__ void k_zero4(float4* p, int n4) {
    int i = blockIdx.x * blockDim.x + threadIdx.x;
    if (i < n4) p[i] = make_float4(0.f, 0.f, 0.f, 0.f);
}

// ---------------- edge pass 1: leaky-relu logits + segment max ----------------
__global__ void k_edge_logits(const int* __restrict__ ei,
                              const float* __restrict__ asrc,
                              const float* __restrict__ adst,
                              float* __restrict__ ework,
                              unsigned* __restrict__ nmax, int H) {
    int i = blockIdx.x * blockDim.x + threadIdx.x;   // i = e*H + hh
    if (i >= EE * H) return;
    int hh = i % H;
    int e  = i / H;
    int s = ei[e];
    int d = ei[EE + e];
    float v = asrc[s * H + hh] + adst[d * H + hh];
    v = v > 0.f ? v : NEG * v;
    ework[i] = v;
    atomicMax(&nmax[d * H + hh], enc_f(v));
}

// ---------------- edge pass 2: exp(e - max) + segment sum ----------------
__global__ void k_edge_exp(const int* __restrict__ ei,
                           float* __restrict__ ework,
                           const unsigned* __restrict__ nmax,
                           float* __restrict__ ndenom, int H) {
    int i = blockIdx.x * blockDim.x + threadIdx.x;
    if (i >= EE * H) return;
    int hh = i % H;
    int e  = i / H;
    int d  = ei[EE + e];
    float ex = expf(ework[i] - dec_f(nmax[d * H + hh]));
    ework[i] = ex;
    atomicAdd(&ndenom[d * H + hh], ex);
}

// ---------------- scatter normalized alpha into dense [dst, src] ----------------
__global__ void k_scatter(const int* __restrict__ ei,
                          const float* __restrict__ ework,
                          const float* __restrict__ ndenom,
                          float* __restrict__ dense, int H, int hh) {
    int e = blockIdx.x * blockDim.x + threadIdx.x;
    if (e >= EE) return;
    int s = ei[e];
    int d = ei[EE + e];
    float a = ework[(size_t)e * H + hh] / (ndenom[d * H + hh] + 1e-16f);
    atomicAdd(&dense[(size_t)d * NN + s], a);
}

// ---------------- elementwise bias (+ optional ELU) ----------------
__global__ void k_bias_elu(float* __restrict__ x, const float* __restrict__ b,
                           int n, int F, int do_elu) {
    int i = blockIdx.x * blockDim.x + threadIdx.x;
    if (i >= n) return;
    int f = i % F;
    float v = x[i] + b[f];
    if (do_elu) v = v > 0.f ? v : (expf(v) - 1.f);
    x[i] = v;
}

// ---------------- fc1: z[b,j] = <out2_flat[b,:], fc1_w[j,:]>, K=102400 ----------
// fc1_w (82 MB) is the only HBM-resident operand in the net -> streamed
// coalesced with prefetch; out2 (3.2 MB) is L2-resident and re-read 200x (once per j).
#define KSPLIT 4
__global__ void k_fc1(const float* __restrict__ out2, const float* __restrict__ w,
                      float* __restrict__ z) {
    const int j  = blockIdx.x / KSPLIT;
    const int ks = blockIdx.x % KSPLIT;
    const int Ktot = ROI * D1;                 // 102400
    const int Kc   = Ktot / KSPLIT;            // 25600
    const int k0   = ks * Kc;
    const float* wp = w + (size_t)j * Ktot;
    float acc[BB];
#pragma unroll
    for (int b = 0; b < BB; ++b) acc[b] = 0.f;
    for (int k = k0 + threadIdx.x; k < k0 + Kc; k += blockDim.x) {
        __builtin_prefetch(&wp[k + 2048], 0, 1);
        float wv = wp[k];
#pragma unroll
        for (int b = 0; b < BB; ++b) acc[b] += wv * out2[(size_t)b * Ktot + k];
    }
    __shared__ float red[BB][256];
#pragma unroll
    for (int b = 0; b < BB; ++b) red[b][threadIdx.x] = acc[b];
    __syncthreads();
    for (int s = 128; s > 0; s >>= 1) {
        if ((int)threadIdx.x < s)
#pragma unroll
            for (int b = 0; b < BB; ++b) red[b][threadIdx.x] += red[b][threadIdx.x + s];
        __syncthreads();
    }
    if (threadIdx.x < BB) atomicAdd(&z[threadIdx.x * ROI + j], red[threadIdx.x][0]);
}

// ---------------- head: +fc1_b, eval BN, ELU, fc2 ----------------
__global__ void k_head(const float* __restrict__ z, const float* __restrict__ fc1_b,
                       const float* __restrict__ bn_g, const float* __restrict__ bn_b,
                       const float* __restrict__ bn_m, const float* __restrict__ bn_v,
                       const float* __restrict__ fc2_w, const float* __restrict__ fc2_b,
                       float* __restrict__ out) {
    __shared__ float zz[BB][ROI];
    for (int i = threadIdx.x; i < BB * ROI; i += blockDim.x) {
        int j = i % ROI;
        float v = z[i] + fc1_b[j];
        v = (v - bn_m[j]) * rsqrtf(bn_v[j] + 1e-5f) * bn_g[j] + bn_b[j];
        v = v > 0.f ? v : (expf(v) - 1.f);
        zz[i / ROI][j] = v;
    }
    __syncthreads();
    if (threadIdx.x < BB * 2) {
        int b = threadIdx.x >> 1;
        int c = threadIdx.x & 1;
        float s = fc2_b[c];
        for (int j = 0; j < ROI; ++j) s += zz[b][j] * fc2_w[c * ROI + j];
        out[b * 2 + c] = s;
    }
}

// ---------------------------------------------------------------------------
extern "C" void kernel_launch(void* const* d_in, const int* in_sizes, int n_in,
                              void* d_out, int out_size, void* d_ws, size_t ws_size,
                              hipStream_t stream) {
    (void)in_sizes; (void)n_in; (void)out_size; (void)ws_size;

    const float* x      = (const float*)d_in[0];
    const int*   ei     = (const int*)  d_in[1];
    const float* W1     = (const float*)d_in[2];
    const float* a1_src = (const float*)d_in[3];
    const float* a1_dst = (const float*)d_in[4];
    const float* b1     = (const float*)d_in[5];
    const float* W2     = (const float*)d_in[6];
    const float* a2_src = (const float*)d_in[7];
    const float* a2_dst = (const float*)d_in[8];
    const float* b2     = (const float*)d_in[9];
    const float* fc1_w  = (const float*)d_in[10];
    const float* fc1_b  = (const float*)d_in[11];
    const float* bn_g   = (const float*)d_in[12];
    const float* bn_b   = (const float*)d_in[13];
    const float* bn_m   = (const float*)d_in[14];
    const float* bn_v   = (const float*)d_in[15];
    const float* fc2_w  = (const float*)d_in[16];
    const float* fc2_b  = (const float*)d_in[17];
    float* out = (float*)d_out;

    // ---- workspace carve-out (~34 MB) ----
    char* ws = (char*)d_ws;
    size_t off = 0;
    auto alloc = [&](size_t bytes) -> void* {
        void* p = ws + off;
        off += (bytes + 255) & ~(size_t)255;
        return p;
    };
    float*    h1     = (float*)   alloc((size_t)NN * D1 * 4);
    float*    out1   = (float*)   alloc((size_t)NN * D1 * 4);
    float*    h2     = (float*)   alloc((size_t)NN * D1 * 4);
    float*    out2   = (float*)   alloc((size_t)NN * D1 * 4);
    float*    ework  = (float*)   alloc((size_t)EE * HIN * 4);
    float*    dense  = (float*)   alloc((size_t)NN * NN * 4);
    float*    asrc   = (float*)   alloc((size_t)NN * HIN * 4);
    float*    adst   = (float*)   alloc((size_t)NN * HIN * 4);
    unsigned* nmax   = (unsigned*)alloc((size_t)NN * HIN * 4);
    float*    ndenom = (float*)   alloc((size_t)NN * HIN * 4);
    float*    z      = (float*)   alloc((size_t)BB * ROI * 4);

    const int T = 256;

    // ================= layer 1 (H=8, F=64) =================
    // h1 = x @ W1   [1600,200]x[200,512]  -> NT=4: grid.x = 512/256 = 2
    k_wmma_gemm_f32<4><<<dim3(2, NN / 16), 128, 0, stream>>>(
        x, W1, h1, NN, D1, ROI, ROI, D1, D1);

    k_attn_coef<<<(NN * HIN + T - 1) / T, T, 0, stream>>>(
        h1, a1_src, a1_dst, asrc, adst, HIN, HID);

    k_zero<<<(NN * HIN + T - 1) / T, T, 0, stream>>>((float*)nmax, NN * HIN);
    k_zero<<<(NN * HIN + T - 1) / T, T, 0, stream>>>(ndenom, NN * HIN);

    k_edge_logits<<<(EE * HIN + T - 1) / T, T, 0, stream>>>(ei, asrc, adst, ework, nmax, HIN);
    k_edge_exp   <<<(EE * HIN + T - 1) / T, T, 0, stream>>>(ei, ework, nmax, ndenom, HIN);

    for (int hh = 0; hh < HIN; ++hh) {
        k_zero4<<<(NN * NN / 4 + T - 1) / T, T, 0, stream>>>((float4*)dense, NN * NN / 4);
        k_scatter<<<(EE + T - 1) / T, T, 0, stream>>>(ei, ework, ndenom, dense, HIN, hh);
        // out1[:, hh*64 : hh*64+64] = dense @ h1[:, hh*64 : hh*64+64]
        // NT=1: 4 waves x 16 cols = 64 cols per block, all waves active
        k_wmma_gemm_f32<1><<<dim3(1, NN / 16), 128, 0, stream>>>(
            dense, h1 + hh * HID, out1 + hh * HID, NN, HID, NN, NN, D1, D1);
    }
    k_bias_elu<<<(NN * D1 + T - 1) / T, T, 0, stream>>>(out1, b1, NN * D1, D1, 1);

    // ================= layer 2 (H=1, F=512) =================
    // h2 = out1 @ W2   [1600,512]x[512,512]
    k_wmma_gemm_f32<4><<<dim3(2, NN / 16), 128, 0, stream>>>(
        out1, W2, h2, NN, D1, D1, D1, D1, D1);

    k_attn_coef<<<(NN + T - 1) / T, T, 0, stream>>>(h2, a2_src, a2_dst, asrc, adst, 1, D1);

    k_zero<<<(NN + T - 1) / T, T, 0, stream>>>((float*)nmax, NN);
    k_zero<<<(NN + T - 1) / T, T, 0, stream>>>(ndenom, NN);

    k_edge_logits<<<(EE + T - 1) / T, T, 0, stream>>>(ei, asrc, adst, ework, nmax, 1);
    k_edge_exp   <<<(EE + T - 1) / T, T, 0, stream>>>(ei, ework, nmax, ndenom, 1);

    k_zero4<<<(NN * NN / 4 + T - 1) / T, T, 0, stream>>>((float4*)dense, NN * NN / 4);
    k_scatter<<<(EE + T - 1) / T, T, 0, stream>>>(ei, ework, ndenom, dense, 1, 0);
    // out2 = dense @ h2   [1600,1600]x[1600,512]
    k_wmma_gemm_f32<4><<<dim3(2, NN / 16), 128, 0, stream>>>(
        dense, h2, out2, NN, D1, NN, NN, D1, D1);
    k_bias_elu<<<(NN * D1 + T - 1) / T, T, 0, stream>>>(out2, b2, NN * D1, D1, 0);

    // ================= classifier head =================
    k_zero<<<(BB * ROI + T - 1) / T, T, 0, stream>>>(z, BB * ROI);
    k_fc1<<<ROI * KSPLIT, 256, 0, stream>>>(out2, fc1_w, z);
    k_head<<<1, 256, 0, stream>>>(z, fc1_b, bn_g, bn_b, bn_m, bn_v, fc2_w, fc2_b, out);
}